// s2v_13597866459920
// MI455X (gfx1250) — compile-verified
//
#include <hip/hip_runtime.h>
#include <hip/hip_bf16.h>

typedef __attribute__((ext_vector_type(2))) float v2f;
typedef __attribute__((ext_vector_type(4))) float v4f;
typedef __attribute__((ext_vector_type(8))) float v8f;

// ---------------------------------------------------------------------------
// Zero workspace (agg_mu [N*64] followed contiguously by degW [N])
// ---------------------------------------------------------------------------
__global__ void s2v_zero_kernel(float* __restrict__ p, int n) {
    int i = blockIdx.x * blockDim.x + threadIdx.x;
    int stride = gridDim.x * blockDim.x;
    for (; i < n; i += stride) p[i] = 0.0f;
}

// ---------------------------------------------------------------------------
// Precompute folded small matrices:
//   rW3[k] = sum_j W3[k,j] * relu(W4[j])
//   v1[p]  = sum_c Wc0[p, c]       * W1[c]
//   v3[p]  = sum_k Wc0[p, 128 + k] * rW3[k]
//   M[p,j] = sum_q Wc0[p, 64 + q]  * W2[q,j]
// One block of 64 threads (thread p owns row p).
// ---------------------------------------------------------------------------
__global__ void s2v_precompute_kernel(const float* __restrict__ W1,
                                      const float* __restrict__ W2,
                                      const float* __restrict__ W3,
                                      const float* __restrict__ W4,
                                      const float* __restrict__ Wc0,
                                      float* __restrict__ M,
                                      float* __restrict__ v1v3) {
    __shared__ float rW3[64];
    int p = threadIdx.x;  // 0..63
    float s = 0.0f;
    for (int j = 0; j < 64; ++j) s += W3[p * 64 + j] * fmaxf(W4[j], 0.0f);
    rW3[p] = s;
    __syncthreads();
    float a = 0.0f, b = 0.0f;
    for (int c = 0; c < 64; ++c) {
        a += Wc0[p * 192 + c] * W1[c];
        b += Wc0[p * 192 + 128 + c] * rW3[c];
    }
    v1v3[p] = a;
    v1v3[64 + p] = b;
    for (int j = 0; j < 64; ++j) {
        float m = 0.0f;
        for (int q = 0; q < 64; ++q) m += Wc0[p * 192 + 64 + q] * W2[q * 64 + j];
        M[p * 64 + j] = m;
    }
}

// ---------------------------------------------------------------------------
// Edge scatter: half-wave (16 lanes) per edge.
//   agg_mu[dst, :] += mu[src, :]   (float4 load, 4 f32 atomics per lane)
//   degW[dst]      += weight[e]    (lane 0 of half-wave)
// mu / agg_mu are 25.6 MB each -> L2 resident; atomics resolve in L2.
// ---------------------------------------------------------------------------
__global__ void s2v_edge_scatter_kernel(const int* __restrict__ ei,
                                        const float* __restrict__ weight,
                                        const float* __restrict__ mu,
                                        float* __restrict__ agg_mu,
                                        float* __restrict__ degW, int E) {
    int tid = blockIdx.x * blockDim.x + threadIdx.x;
    int l16 = threadIdx.x & 15;
    int hw = tid >> 4;
    int nHw = (gridDim.x * blockDim.x) >> 4;
    for (int e = hw; e < E; e += nHw) {
        int src = ei[e];
        int dst = ei[E + e];
        v4f m = *(const v4f*)(mu + (size_t)src * 64 + l16 * 4);
        float* dp = agg_mu + (size_t)dst * 64 + l16 * 4;
        atomicAdd(dp + 0, m.x);
        atomicAdd(dp + 1, m.y);
        atomicAdd(dp + 2, m.z);
        atomicAdd(dp + 3, m.w);
        if (l16 == 0) atomicAdd(degW + dst, weight[e]);
    }
}

// ---------------------------------------------------------------------------
// Node transform: one wave per 16-node tile.
//   h1  = relu(agg_mu @ M^T + x*v1 + degW*v3)     (WMMA f32 16x16x4)
//   out = relu(h1 @ Wc1^T)                        (WMMA f32 16x16x4)
// M, Wc1 staged in LDS as B operands; per-wave LDS stage does the D->A layout
// transform between the two GEMMs and the coalesced output store.
// blockDim must be 128 (4 waves); loop bounds wave-uniform => EXEC all-ones
// at every WMMA (ISA requirement).
// ---------------------------------------------------------------------------
__global__ void s2v_node_kernel(const float* __restrict__ agg_mu,
                                const float* __restrict__ x,
                                const float* __restrict__ degW,
                                const float* __restrict__ M,
                                const float* __restrict__ Wc1,
                                const float* __restrict__ v1v3,
                                float* __restrict__ out, int numTiles) {
    __shared__ float sM[64 * 64];
    __shared__ float sW[64 * 64];
    __shared__ float sV[128];
    __shared__ float stage[4][16 * 64];  // per-wave staging (4 KB each)

    int tid = threadIdx.x;
    for (int i = tid; i < 4096; i += blockDim.x) {
        sM[i] = M[i];
        sW[i] = Wc1[i];
    }
    for (int i = tid; i < 128; i += blockDim.x) sV[i] = v1v3[i];
    __syncthreads();

    int wave = tid >> 5;       // 0..3
    int lane = tid & 31;
    int lh = lane & 15;        // lane % 16
    int hi = lane >> 4;        // 0 or 1 (which half of the wave)
    float* st = &stage[wave][0];

    int gw = blockIdx.x * 4 + wave;
    int nw = gridDim.x * 4;

    for (int tile = gw; tile < numTiles; tile += nw) {
        int nodeBase = tile * 16;

        // --- A fragments for GEMM1: A[m=lh, k=4*ks + 2*hi + {0,1}] ---
        v2f afrag[16];
        const float* arow = agg_mu + (size_t)(nodeBase + lh) * 64 + hi * 2;
        #pragma unroll
        for (int k = 0; k < 16; ++k) afrag[k] = *(const v2f*)(arow + k * 4);

        // per-lane bias rows (C/D layout: row = r + 8*hi)
        float xb[8], db[8];
        #pragma unroll
        for (int r = 0; r < 8; ++r) {
            int node = nodeBase + r + hi * 8;
            xb[r] = x[node];
            db[r] = degW[node];
        }

        // --- GEMM1: T = A @ M^T  (4 col tiles x 16 k-steps) ---
        #pragma unroll
        for (int t = 0; t < 4; ++t) {
            v8f c = {};
            #pragma unroll
            for (int k = 0; k < 16; ++k) {
                int kb = k * 4 + hi * 2;
                v2f b = *(const v2f*)&sM[(t * 16 + lh) * 64 + kb];
                c = __builtin_amdgcn_wmma_f32_16x16x4_f32(
                    false, afrag[k], false, b, (short)0, c, false, false);
            }
            // bias + relu, stage h1 into LDS (row-major [node_local][p])
            int p = t * 16 + lh;
            float v1p = sV[p], v3p = sV[64 + p];
            #pragma unroll
            for (int r = 0; r < 8; ++r) {
                float h = c[r] + xb[r] * v1p + db[r] * v3p;
                st[(r + hi * 8) * 64 + p] = fmaxf(h, 0.0f);
            }
        }
        // same-wave DS ordering + compiler barrier before re-reading stage
        asm volatile("s_wait_dscnt 0" ::: "memory");

        // --- A fragments for GEMM2 from staged h1 ---
        v2f a2[16];
        #pragma unroll
        for (int k = 0; k < 16; ++k)
            a2[k] = *(const v2f*)&st[lh * 64 + k * 4 + hi * 2];

        // --- GEMM2: out = relu(h1 @ Wc1^T), restage into LDS ---
        #pragma unroll
        for (int t = 0; t < 4; ++t) {
            v8f c = {};
            #pragma unroll
            for (int k = 0; k < 16; ++k) {
                int kb = k * 4 + hi * 2;
                v2f b = *(const v2f*)&sW[(t * 16 + lh) * 64 + kb];
                c = __builtin_amdgcn_wmma_f32_16x16x4_f32(
                    false, a2[k], false, b, (short)0, c, false, false);
            }
            #pragma unroll
            for (int r = 0; r < 8; ++r)
                st[(r + hi * 8) * 64 + t * 16 + lh] = fmaxf(c[r], 0.0f);
        }
        asm volatile("s_wait_dscnt 0" ::: "memory");

        // --- coalesced store: 16 rows x 64 floats = 1024 floats, float4 ---
        float* op = out + (size_t)nodeBase * 64;
        #pragma unroll
        for (int j = 0; j < 8; ++j) {
            v4f v = *(const v4f*)&st[(j * 32 + lane) * 4];
            *(v4f*)(op + (j * 32 + lane) * 4) = v;
        }
    }
}

// ---------------------------------------------------------------------------
// Launch
// inputs: 0:x[N] 1:mu[N*64] 2:weight[E] 3:edge_index[2*E] 4:W1[64] 5:W2[4096]
//         6:W3[4096] 7:W4[64] 8:Wc0[12288] 9:Wc1[4096]  -> out [N*64] f32
// ---------------------------------------------------------------------------
extern "C" void kernel_launch(void* const* d_in, const int* in_sizes, int n_in,
                              void* d_out, int out_size, void* d_ws, size_t ws_size,
                              hipStream_t stream) {
    const float* x      = (const float*)d_in[0];
    const float* mu     = (const float*)d_in[1];
    const float* weight = (const float*)d_in[2];
    const int*   ei     = (const int*)d_in[3];
    const float* W1     = (const float*)d_in[4];
    const float* W2     = (const float*)d_in[5];
    const float* W3     = (const float*)d_in[6];
    const float* W4     = (const float*)d_in[7];
    const float* Wc0    = (const float*)d_in[8];
    const float* Wc1    = (const float*)d_in[9];
    float* out = (float*)d_out;

    int N = in_sizes[0];
    int E = in_sizes[2];

    float* ws     = (float*)d_ws;
    float* agg_mu = ws;                         // N*64
    float* degW   = ws + (size_t)N * 64;        // N   (contiguous after agg_mu)
    float* Mmat   = degW + N;                   // 4096
    float* v1v3   = Mmat + 4096;                // 128

    // zero agg_mu + degW in one pass (contiguous N*65 floats)
    s2v_zero_kernel<<<1024, 256, 0, stream>>>(agg_mu, N * 65);

    s2v_precompute_kernel<<<1, 64, 0, stream>>>(W1, W2, W3, W4, Wc0, Mmat, v1v3);

    s2v_edge_scatter_kernel<<<8192, 256, 0, stream>>>(ei, weight, mu, agg_mu, degW, E);

    int tiles = N / 16;                         // N = 100000 -> 6250 tiles
    int blocks = (tiles + 3) / 4;               // 4 waves per block, 1 tile/wave
    s2v_node_kernel<<<blocks, 128, 0, stream>>>(agg_mu, x, degW, Mmat, Wc1, v1v3,
                                                out, tiles);
}